// MaskEncoder_44736379355436
// MI455X (gfx1250) — compile-verified
//
#include <hip/hip_runtime.h>
#include <hip/hip_bf16.h>
#include <math.h>

// ---------------- problem constants (from reference) ----------------
#define GAT_N       50000
#define GAT_E       800000
#define GAT_EAUG    (GAT_E + GAT_N)   // with self loops
#define IN_DIM      256
#define H_DIM       64
#define OUT_DIM     16
#define NEG_SLOPE   0.2f
#define GAT_EPS     1e-16f
#define K_HOMO      760000            // int(E*0.95)
#define K_HET       40000             // int(E*0.05)
#define SORT_M      (1 << 20)         // pow2 pad for bitonic sort of E keys

typedef __attribute__((ext_vector_type(2))) float v2f;
typedef __attribute__((ext_vector_type(8))) float v8f;
typedef __attribute__((ext_vector_type(4))) int   v4i;

// ---- CDNA5 async global->LDS path (guarded: falls back to sync fill) ----
#if defined(__gfx1250__) && __has_builtin(__builtin_amdgcn_global_load_async_to_lds_b128)
#define HAVE_ASYNC_LDS 1
typedef __attribute__((address_space(1))) v4i glob_v4i;   // global source
typedef __attribute__((address_space(3))) v4i lds_v4i;    // LDS destination
#define ASYNC_B128(g, l) \
    __builtin_amdgcn_global_load_async_to_lds_b128((glob_v4i*)(g), (lds_v4i*)(l), 0, 0)
#if __has_builtin(__builtin_amdgcn_s_wait_asynccnt)
#define WAIT_ASYNC(N) __builtin_amdgcn_s_wait_asynccnt(N)
#else
#define WAIT_ASYNC(N) asm volatile("s_wait_asynccnt " #N)
#endif
#else
#define HAVE_ASYNC_LDS 0
#endif

// monotone float->uint mapping so unsigned atomicMax == float max
__device__ __forceinline__ unsigned enc_f32(float f) {
    unsigned b = __float_as_uint(f);
    return (b & 0x80000000u) ? ~b : (b | 0x80000000u);
}
__device__ __forceinline__ float dec_f32(unsigned k) {
    unsigned b = (k & 0x80000000u) ? (k ^ 0x80000000u) : ~k;
    return __uint_as_float(b);
}
__device__ __forceinline__ float leaky(float e) {
    return e > 0.0f ? e : NEG_SLOPE * e;
}

// ---------------- init: zero agg, reset max-keys / denom ----------------
__global__ void init_layer_kernel(unsigned* __restrict__ mkey, float* __restrict__ denom,
                                  float* __restrict__ agg, int n, int F) {
    int i = blockIdx.x * blockDim.x + threadIdx.x;
    if (i < n * F) agg[i] = 0.0f;
    if (i < n) { mkey[i] = 0u; denom[i] = 0.0f; }   // key 0 encodes "-inf-ish"
}

// ---------------- layer-1 GEMM: g1[N,64] = x[N,256] @ W1[256,64] ----------------
// 8 waves/block, wave handles 16 rows x 64 cols via 4 WMMA accumulators.
// W tiles double-buffered in LDS via async global->LDS copies (ASYNCcnt).
__global__ void __launch_bounds__(256) gemm1_wmma_kernel(const float* __restrict__ x,
                                                         const float* __restrict__ W,
                                                         float* __restrict__ g, int n) {
    __shared__ float sW[2][64][65];                 // double-buffered K-chunk x 64 cols, padded
    const int lane = threadIdx.x & 31;
    const int wave = threadIdx.x >> 5;
    const int row0 = blockIdx.x * 128 + wave * 16;
    const int mA   = lane & 15;                     // A-frag row
    const int kb   = (lane >> 4) * 2;               // A/B-frag k sub-offset
    const int nB   = lane & 15;                     // B/C-frag col
    v8f acc0 = {}, acc1 = {}, acc2 = {}, acc3 = {};

    int rowc = row0 + mA; if (rowc >= n) rowc = n - 1;   // clamp (padded rows never stored)
    const float* xr = x + (size_t)rowc * IN_DIM + kb;

#if HAVE_ASYNC_LDS
    // per-thread: 4 x B128 copies per 64x64 chunk; copies stay within one 64-float row,
    // so the padded (65-stride) LDS layout is preserved.
    {
        #pragma unroll
        for (int it = 0; it < 4; ++it) {
            int c = it * 256 + threadIdx.x;         // 0..1023 b128 chunks
            int row = c >> 4, col = (c & 15) * 4;
            ASYNC_B128(W + (size_t)row * H_DIM + col, &sW[0][row][col]);
        }
    }
#endif

    for (int kc = 0; kc < 4; ++kc) {
        const int buf = kc & 1;
#if HAVE_ASYNC_LDS
        if (kc < 3) {                                // prefetch next chunk into other buffer
            #pragma unroll
            for (int it = 0; it < 4; ++it) {
                int c = it * 256 + threadIdx.x;
                int row = c >> 4, col = (c & 15) * 4;
                ASYNC_B128(W + (size_t)((kc + 1) * 64 + row) * H_DIM + col,
                           &sW[buf ^ 1][row][col]);
            }
            WAIT_ASYNC(4);                           // in-order: current chunk complete
        } else {
            WAIT_ASYNC(0);
        }
#else
        for (int t = threadIdx.x; t < 64 * 64; t += 256) {
            int kk = t >> 6, nn = t & 63;
            sW[buf][kk][nn] = W[(size_t)(kc * 64 + kk) * H_DIM + nn];
        }
#endif
        __syncthreads();
        const float* xc = xr + kc * 64;
        if (kc < 3) __builtin_prefetch(xc + 64, 0, 1);  // global_prefetch_b8: next A chunk
        #pragma unroll
        for (int kk = 0; kk < 64; kk += 4) {
            v2f a; a.x = xc[kk]; a.y = xc[kk + 1];
            v2f b0, b1, b2, b3;
            b0.x = sW[buf][kk + kb][nB];      b0.y = sW[buf][kk + kb + 1][nB];
            b1.x = sW[buf][kk + kb][16 + nB]; b1.y = sW[buf][kk + kb + 1][16 + nB];
            b2.x = sW[buf][kk + kb][32 + nB]; b2.y = sW[buf][kk + kb + 1][32 + nB];
            b3.x = sW[buf][kk + kb][48 + nB]; b3.y = sW[buf][kk + kb + 1][48 + nB];
            acc0 = __builtin_amdgcn_wmma_f32_16x16x4_f32(false, a, false, b0, (short)0, acc0, false, false);
            acc1 = __builtin_amdgcn_wmma_f32_16x16x4_f32(false, a, false, b1, (short)0, acc1, false, false);
            acc2 = __builtin_amdgcn_wmma_f32_16x16x4_f32(false, a, false, b2, (short)0, acc2, false, false);
            acc3 = __builtin_amdgcn_wmma_f32_16x16x4_f32(false, a, false, b3, (short)0, acc3, false, false);
        }
        __syncthreads();                             // all reads done before buffer reuse
    }
    const int mBase = (lane >> 4) * 8;              // C/D layout: VGPR r -> M=r / M=r+8
    #pragma unroll
    for (int r = 0; r < 8; ++r) {
        int row = row0 + mBase + r;
        if (row < n) {
            float* gr = g + (size_t)row * H_DIM + nB;
            gr[0] = acc0[r]; gr[16] = acc1[r]; gr[32] = acc2[r]; gr[48] = acc3[r];
        }
    }
}

// ---------------- layer-2 GEMM: g2[N,16] = h1[N,64] @ W2[64,16] ----------------
__global__ void __launch_bounds__(256) gemm2_wmma_kernel(const float* __restrict__ h1,
                                                         const float* __restrict__ W,
                                                         float* __restrict__ g, int n) {
    __shared__ float sW[64][17];
    const int lane = threadIdx.x & 31;
    const int wave = threadIdx.x >> 5;
    const int row0 = blockIdx.x * 128 + wave * 16;
    const int mA   = lane & 15;
    const int kb   = (lane >> 4) * 2;
    const int nB   = lane & 15;
    v8f acc = {};

#if HAVE_ASYNC_LDS
    {   // 64 rows x 16 floats = 256 B128 chunks, one per thread (row-preserving => padding OK)
        int row = threadIdx.x >> 2, col = (threadIdx.x & 3) * 4;
        ASYNC_B128(W + (size_t)row * OUT_DIM + col, &sW[row][col]);
        WAIT_ASYNC(0);
    }
#else
    for (int t = threadIdx.x; t < 64 * 16; t += 256) {
        int kk = t >> 4, nn = t & 15;
        sW[kk][nn] = W[kk * OUT_DIM + nn];
    }
#endif
    __syncthreads();

    int rowc = row0 + mA; if (rowc >= n) rowc = n - 1;
    const float* hr = h1 + (size_t)rowc * H_DIM + kb;
    #pragma unroll
    for (int kk = 0; kk < 64; kk += 4) {
        v2f a; a.x = hr[kk]; a.y = hr[kk + 1];
        v2f b; b.x = sW[kk + kb][nB]; b.y = sW[kk + kb + 1][nB];
        acc = __builtin_amdgcn_wmma_f32_16x16x4_f32(false, a, false, b, (short)0, acc, false, false);
    }
    const int mBase = (lane >> 4) * 8;
    #pragma unroll
    for (int r = 0; r < 8; ++r) {
        int row = row0 + mBase + r;
        if (row < n) g[(size_t)row * OUT_DIM + nB] = acc[r];
    }
}

// ---------------- per-node attention coefficients ----------------
__global__ void node_alpha_kernel(const float* __restrict__ g, const float* __restrict__ a_src,
                                  const float* __restrict__ a_dst, float* __restrict__ as_,
                                  float* __restrict__ ad_, int n, int F) {
    int i = blockIdx.x * blockDim.x + threadIdx.x;
    if (i >= n) return;
    const float* gi = g + (size_t)i * F;
    float s = 0.0f, d = 0.0f;
    for (int f = 0; f < F; ++f) { float v = gi[f]; s += v * a_src[f]; d += v * a_dst[f]; }
    as_[i] = s; ad_[i] = d;
}

// ---------------- edge pass A: segment max ----------------
__global__ void edge_max_kernel(const int* __restrict__ src, const int* __restrict__ dst,
                                const float* __restrict__ as_, const float* __restrict__ ad_,
                                unsigned* __restrict__ mkey) {
    int e = blockIdx.x * blockDim.x + threadIdx.x;
    if (e >= GAT_EAUG) return;
    int s, d;
    if (e < GAT_E) { s = src[e]; d = dst[e]; } else { s = d = e - GAT_E; }
    atomicMax(&mkey[d], enc_f32(leaky(as_[s] + ad_[d])));
}

// ---------------- edge pass B: exp + segment sum ----------------
__global__ void edge_exp_kernel(const int* __restrict__ src, const int* __restrict__ dst,
                                const float* __restrict__ as_, const float* __restrict__ ad_,
                                const unsigned* __restrict__ mkey, float* __restrict__ p,
                                float* __restrict__ denom) {
    int e = blockIdx.x * blockDim.x + threadIdx.x;
    if (e >= GAT_EAUG) return;
    int s, d;
    if (e < GAT_E) { s = src[e]; d = dst[e]; } else { s = d = e - GAT_E; }
    float pe = __expf(leaky(as_[s] + ad_[d]) - dec_f32(mkey[d]));
    p[e] = pe;
    atomicAdd(&denom[d], pe);
}

// ---------------- edge pass C: weighted scatter-add, one wave per edge ----------------
__global__ void edge_agg_kernel(const int* __restrict__ src, const int* __restrict__ dst,
                                const float* __restrict__ p, const float* __restrict__ denom,
                                const float* __restrict__ g, float* __restrict__ agg, int F) {
    int wave = blockIdx.x * (blockDim.x >> 5) + (threadIdx.x >> 5);
    int lane = threadIdx.x & 31;
    if (wave >= GAT_EAUG) return;
    int s, d;
    if (wave < GAT_E) { s = src[wave]; d = dst[wave]; } else { s = d = wave - GAT_E; }
    float w = p[wave] / (denom[d] + GAT_EPS);
    const float* gs = g + (size_t)s * F;
    float* ag = agg + (size_t)d * F;
    for (int f = lane; f < F; f += 32) atomicAdd(&ag[f], w * gs[f]);
}

// ---------------- bias (+relu) epilogue, in place ----------------
__global__ void finalize_kernel(float* __restrict__ h, const float* __restrict__ b,
                                int n, int F, int do_relu) {
    int i = blockIdx.x * blockDim.x + threadIdx.x;
    if (i >= n * F) return;
    float v = h[i] + b[i % F];
    if (do_relu) v = fmaxf(v, 0.0f);
    h[i] = v;
}

// ---------------- edge scores -> packed sort keys (desc sort, stable by idx) ----------------
__global__ void build_keys_kernel(const int* __restrict__ src, const int* __restrict__ dst,
                                  const float* __restrict__ h2, unsigned long long* __restrict__ keys) {
    int i = blockIdx.x * blockDim.x + threadIdx.x;
    if (i >= SORT_M) return;
    unsigned long long key = 0ull;                  // pad: sorts to the end
    if (i < GAT_E) {
        const float* a = h2 + (size_t)src[i] * OUT_DIM;
        const float* b = h2 + (size_t)dst[i] * OUT_DIM;
        float v = 0.0f;
        #pragma unroll
        for (int f = 0; f < OUT_DIM; ++f) v += a[f] * b[f];
        key = ((unsigned long long)enc_f32(v) << 32) |
              (unsigned long long)(0xFFFFFFFFu - (unsigned)i);   // ties: smaller idx first
    }
    keys[i] = key;
}

// ---------------- global bitonic sort step (descending) ----------------
__global__ void bitonic_step_kernel(unsigned long long* __restrict__ d, unsigned j, unsigned k) {
    unsigned i = blockIdx.x * blockDim.x + threadIdx.x;
    unsigned ixj = i ^ j;
    if (ixj > i && i < SORT_M) {
        unsigned long long a = d[i], b = d[ixj];
        bool desc = ((i & k) == 0);
        if (desc ? (a < b) : (a > b)) { d[i] = b; d[ixj] = a; }
    }
}

// ---------------- gather selected edges to output ----------------
__global__ void write_out_kernel(const unsigned long long* __restrict__ keys,
                                 const int* __restrict__ src, const int* __restrict__ dst,
                                 int* __restrict__ out) {
    int i = blockIdx.x * blockDim.x + threadIdx.x;
    if (i < K_HOMO) {
        unsigned idx = 0xFFFFFFFFu - (unsigned)(keys[i] & 0xFFFFFFFFull);
        out[i] = src[idx];
        out[K_HOMO + i] = dst[idx];
    } else if (i < K_HOMO + K_HET) {
        int r = i - K_HOMO;                                   // ascending value order
        unsigned idx = 0xFFFFFFFFu - (unsigned)(keys[GAT_E - 1 - r] & 0xFFFFFFFFull);
        out[2 * K_HOMO + r] = src[idx];
        out[2 * K_HOMO + K_HET + r] = dst[idx];
    }
}

// ==================================================================
extern "C" void kernel_launch(void* const* d_in, const int* in_sizes, int n_in,
                              void* d_out, int out_size, void* d_ws, size_t ws_size,
                              hipStream_t stream) {
    const float* x      = (const float*)d_in[0];
    const int*   eidx   = (const int*)d_in[1];
    const float* W1     = (const float*)d_in[2];
    const float* a1s    = (const float*)d_in[3];
    const float* a1d    = (const float*)d_in[4];
    const float* b1     = (const float*)d_in[5];
    const float* W2     = (const float*)d_in[6];
    const float* a2s    = (const float*)d_in[7];
    const float* a2d    = (const float*)d_in[8];
    const float* b2     = (const float*)d_in[9];
    const int* src = eidx;            // edge_index row 0
    const int* dst = eidx + GAT_E;    // edge_index row 1
    int* out = (int*)d_out;

    // ---- bump-allocate workspace ----
    char* ws = (char*)d_ws; size_t off = 0;
    auto alloc = [&](size_t bytes) { void* p = ws + off; off = (off + bytes + 255) & ~(size_t)255; return p; };
    float*    g1     = (float*)alloc((size_t)GAT_N * H_DIM * 4);
    float*    as1    = (float*)alloc((size_t)GAT_N * 4);
    float*    ad1    = (float*)alloc((size_t)GAT_N * 4);
    unsigned* mk1    = (unsigned*)alloc((size_t)GAT_N * 4);
    float*    den1   = (float*)alloc((size_t)GAT_N * 4);
    float*    agg1   = (float*)alloc((size_t)GAT_N * H_DIM * 4);   // becomes h1 in place
    float*    p1     = (float*)alloc((size_t)GAT_EAUG * 4);
    float*    g2     = (float*)alloc((size_t)GAT_N * OUT_DIM * 4);
    float*    as2    = (float*)alloc((size_t)GAT_N * 4);
    float*    ad2    = (float*)alloc((size_t)GAT_N * 4);
    unsigned* mk2    = (unsigned*)alloc((size_t)GAT_N * 4);
    float*    den2   = (float*)alloc((size_t)GAT_N * 4);
    float*    agg2   = (float*)alloc((size_t)GAT_N * OUT_DIM * 4); // becomes h2 in place
    float*    p2     = (float*)alloc((size_t)GAT_EAUG * 4);
    unsigned long long* keys = (unsigned long long*)alloc((size_t)SORT_M * 8);
    (void)ws_size; (void)in_sizes; (void)n_in; (void)out_size;

    const int TB = 256;
    const int gRows  = (GAT_N + 127) / 128;                       // 8 waves x 16 rows
    const int gNode  = (GAT_N + TB - 1) / TB;
    const int gEdge  = (GAT_EAUG + TB - 1) / TB;
    const int gWaveE = (GAT_EAUG + (TB / 32) - 1) / (TB / 32);

    // ---------- layer 1 ----------
    init_layer_kernel<<<(GAT_N * H_DIM + TB - 1) / TB, TB, 0, stream>>>(mk1, den1, agg1, GAT_N, H_DIM);
    gemm1_wmma_kernel<<<gRows, TB, 0, stream>>>(x, W1, g1, GAT_N);
    node_alpha_kernel<<<gNode, TB, 0, stream>>>(g1, a1s, a1d, as1, ad1, GAT_N, H_DIM);
    edge_max_kernel<<<gEdge, TB, 0, stream>>>(src, dst, as1, ad1, mk1);
    edge_exp_kernel<<<gEdge, TB, 0, stream>>>(src, dst, as1, ad1, mk1, p1, den1);
    edge_agg_kernel<<<gWaveE, TB, 0, stream>>>(src, dst, p1, den1, g1, agg1, H_DIM);
    finalize_kernel<<<(GAT_N * H_DIM + TB - 1) / TB, TB, 0, stream>>>(agg1, b1, GAT_N, H_DIM, 1);

    // ---------- layer 2 ----------
    init_layer_kernel<<<(GAT_N * OUT_DIM + TB - 1) / TB, TB, 0, stream>>>(mk2, den2, agg2, GAT_N, OUT_DIM);
    gemm2_wmma_kernel<<<gRows, TB, 0, stream>>>(agg1 /*h1*/, W2, g2, GAT_N);
    node_alpha_kernel<<<gNode, TB, 0, stream>>>(g2, a2s, a2d, as2, ad2, GAT_N, OUT_DIM);
    edge_max_kernel<<<gEdge, TB, 0, stream>>>(src, dst, as2, ad2, mk2);
    edge_exp_kernel<<<gEdge, TB, 0, stream>>>(src, dst, as2, ad2, mk2, p2, den2);
    edge_agg_kernel<<<gWaveE, TB, 0, stream>>>(src, dst, p2, den2, g2, agg2, OUT_DIM);
    finalize_kernel<<<(GAT_N * OUT_DIM + TB - 1) / TB, TB, 0, stream>>>(agg2, b2, GAT_N, OUT_DIM, 0);

    // ---------- edge scores + full descending sort (bitonic, 2^20 padded) ----------
    build_keys_kernel<<<SORT_M / TB, TB, 0, stream>>>(src, dst, agg2 /*h2*/, keys);
    for (unsigned k = 2; k <= SORT_M; k <<= 1)
        for (unsigned j = k >> 1; j > 0; j >>= 1)
            bitonic_step_kernel<<<SORT_M / TB, TB, 0, stream>>>(keys, j, k);

    // ---------- top-k selection -> output ----------
    write_out_kernel<<<(K_HOMO + K_HET + TB - 1) / TB, TB, 0, stream>>>(keys, src, dst, out);
}